// HashEmbedder_4123168604709
// MI455X (gfx1250) — compile-verified
//
#include <hip/hip_runtime.h>
#include <math.h>

#define NUM_LEVELS 16
#define LEVEL_DIM  2
#define LOG2_T     21
#define TBL        (1u << LOG2_T)
#define HMASK      (TBL - 1u)
#define BLOCK      256

typedef float f2_t __attribute__((ext_vector_type(2)));

struct ResArr { float r[NUM_LEVELS]; };

__global__ __launch_bounds__(BLOCK) void hashgrid_fwd(
    const float* __restrict__ x,
    const float* __restrict__ tables,
    float* __restrict__ out,
    ResArr res, int n_points)
{
    const int level = (int)blockIdx.y;                       // uniform per block -> SGPR
    const int n = (int)(blockIdx.x * BLOCK + threadIdx.x);
    if (n >= n_points) return;

    const float rf = res.r[level];                           // scalar load from kernarg
    const f2_t* __restrict__ tb =
        reinterpret_cast<const f2_t*>(tables) + (size_t)level * TBL;  // SGPR base

    // ---- load point, normalize to [0,1] ----
    const size_t xb = 3u * (size_t)n;
    const float px = x[xb + 0];
    const float py = x[xb + 1];
    const float pz = x[xb + 2];

    float nx = fminf(fmaxf((px + 2.0f) * 0.25f, 0.0f), 1.0f);
    float ny = fminf(fmaxf((py + 2.0f) * 0.25f, 0.0f), 1.0f);
    float nz = fminf(fmaxf((pz + 2.0f) * 0.25f, 0.0f), 1.0f);

    const float sx = nx * rf, sy = ny * rf, sz = nz * rf;
    const float fx = floorf(sx), fy = floorf(sy), fz = floorf(sz);
    const float wx = sx - fx, wy = sy - fy, wz = sz - fz;

    const unsigned ix = (unsigned)(int)fx;
    const unsigned iy = (unsigned)(int)fy;
    const unsigned iz = (unsigned)(int)fz;

    // ---- spatial hash of the 8 cube corners (uint32 wrap == ref's masked int32) ----
    const unsigned P0 = 73856093u, P1 = 19349663u, P2 = 83492791u;
    const unsigned ax0 = ix * P0, ax1 = ax0 + P0;
    const unsigned ay0 = iy * P1, ay1 = ay0 + P1;
    const unsigned az0 = iz * P2, az1 = az0 + P2;

    const unsigned t00 = ax0 ^ ay0, t01 = ax0 ^ ay1;
    const unsigned t10 = ax1 ^ ay0, t11 = ax1 ^ ay1;

    const unsigned h000 = (t00 ^ az0) & HMASK;
    const unsigned h001 = (t00 ^ az1) & HMASK;
    const unsigned h010 = (t01 ^ az0) & HMASK;
    const unsigned h011 = (t01 ^ az1) & HMASK;
    const unsigned h100 = (t10 ^ az0) & HMASK;
    const unsigned h101 = (t10 ^ az1) & HMASK;
    const unsigned h110 = (t11 ^ az0) & HMASK;
    const unsigned h111 = (t11 ^ az1) & HMASK;

    // ---- 8 independent b64 gathers: all issued before first use (MLP) ----
    const f2_t e000 = tb[h000];
    const f2_t e001 = tb[h001];
    const f2_t e010 = tb[h010];
    const f2_t e011 = tb[h011];
    const f2_t e100 = tb[h100];
    const f2_t e101 = tb[h101];
    const f2_t e110 = tb[h110];
    const f2_t e111 = tb[h111];

    // ---- trilinear weights: prod over dims of (off ? w : 1-w), corner order 000..111 ----
    const float ux = 1.0f - wx, uy = 1.0f - wy, uz = 1.0f - wz;
    const float w000 = (ux * uy) * uz;
    const float w001 = (ux * uy) * wz;
    const float w010 = (ux * wy) * uz;
    const float w011 = (ux * wy) * wz;
    const float w100 = (wx * uy) * uz;
    const float w101 = (wx * uy) * wz;
    const float w110 = (wx * wy) * uz;
    const float w111 = (wx * wy) * wz;

    float ox = w000 * e000.x;
    float oy = w000 * e000.y;
    ox = fmaf(w001, e001.x, ox);  oy = fmaf(w001, e001.y, oy);
    ox = fmaf(w010, e010.x, ox);  oy = fmaf(w010, e010.y, oy);
    ox = fmaf(w011, e011.x, ox);  oy = fmaf(w011, e011.y, oy);
    ox = fmaf(w100, e100.x, ox);  oy = fmaf(w100, e100.y, oy);
    ox = fmaf(w101, e101.x, ox);  oy = fmaf(w101, e101.y, oy);
    ox = fmaf(w110, e110.x, ox);  oy = fmaf(w110, e110.y, oy);
    ox = fmaf(w111, e111.x, ox);  oy = fmaf(w111, e111.y, oy);

    // ---- out[n, level*2 + d]: 8B-aligned vector store, non-temporal so the
    // write-once output stream doesn't evict hash-table lines from L2 ----
    f2_t o; o.x = ox; o.y = oy;
    f2_t* __restrict__ op = reinterpret_cast<f2_t*>(
        out + (size_t)n * (NUM_LEVELS * LEVEL_DIM) + level * LEVEL_DIM);
    __builtin_nontemporal_store(o, op);
}

extern "C" void kernel_launch(void* const* d_in, const int* in_sizes, int n_in,
                              void* d_out, int out_size, void* d_ws, size_t ws_size,
                              hipStream_t stream) {
    const float* x      = (const float*)d_in[0];
    const float* tables = (const float*)d_in[1];
    float* out          = (float*)d_out;

    const int n_points = in_sizes[0] / 3;

    // Replicate the reference's float64 resolution computation exactly:
    // b = exp((ln 2048 - ln 16)/15); res_i = float(int(floor(16 * b**i)))
    ResArr res;
    const double b = exp((log(2048.0) - log(16.0)) / 15.0);
    for (int i = 0; i < NUM_LEVELS; ++i)
        res.r[i] = (float)floor(16.0 * pow(b, (double)i));

    dim3 grid((unsigned)((n_points + BLOCK - 1) / BLOCK), NUM_LEVELS, 1);
    hashgrid_fwd<<<grid, dim3(BLOCK, 1, 1), 0, stream>>>(x, tables, out, res, n_points);
}